// Differ_Amplifier_33071248179821
// MI455X (gfx1250) — compile-verified
//
#include <hip/hip_runtime.h>
#include <math.h>

typedef __attribute__((ext_vector_type(2))) float v2f;
typedef __attribute__((ext_vector_type(8))) float v8f;

#define HID 512
#define NROWS 32768
#define LAYERS 4
#define CS_BLOCKS 256
#define CS_ROWS (NROWS / CS_BLOCKS) /* 128 rows per partial-sum block */
#define LDS_STRIDE 516              /* 16B (4-bank) row skew -> conflict-free frag loads */

// ---------------------------------------------------------------------------
// Stage 1: deterministic two-level column-sum.  partial[b][j] = sum of 128 rows.
// Fully coalesced (each row of 512 floats read by 256 threads as 2 passes).
// ---------------------------------------------------------------------------
__global__ __launch_bounds__(256) void colsum_partial_kernel(
    const float* __restrict__ h, float* __restrict__ partial) {
  const int b = blockIdx.x;
  const int t = threadIdx.x;
  const float* base = h + (size_t)b * CS_ROWS * HID;
  float s0 = 0.0f, s1 = 0.0f;
  for (int r = 0; r < CS_ROWS; ++r) {
    s0 += base[(size_t)r * HID + t];
    s1 += base[(size_t)r * HID + t + 256];
  }
  partial[(size_t)b * HID + t]       = s0;
  partial[(size_t)b * HID + t + 256] = s1;
}

// ---------------------------------------------------------------------------
// Stage 2: total[j] = sum_p partial[p][j];  t[o] = sum_j total[j] * W[o][j]
// Single block, barrier between phases -> deterministic.
// ---------------------------------------------------------------------------
__global__ __launch_bounds__(512) void reduce_t_kernel(
    const float* __restrict__ partial, const float* __restrict__ W,
    float* __restrict__ t_out) {
  __shared__ float total[HID];
  const int j = threadIdx.x;
  float s = 0.0f;
  for (int p = 0; p < CS_BLOCKS; ++p) s += partial[(size_t)p * HID + j];
  total[j] = s;
  __syncthreads();
  const float* wr = W + (size_t)j * HID;  // row o = j of W (out-major)
  float acc = 0.0f;
  for (int k = 0; k < HID; ++k) acc += total[k] * wr[k];
  t_out[j] = acc;
}

// ---------------------------------------------------------------------------
// Stage 3: h_out[n,o] = h_in[n,o] + (N * (h_in @ W^T)[n,o] - t[o]) / (N-1)
// Wave tile: 16 (M) x 64 (O), 4 x v8f accumulators, K stepped by 4 with
// v_wmma_f32_16x16x4_f32.  Block = 8 waves stacked in M (128 x 64 tile).
// h (64MB) and W (1MB) are L2-resident on MI455X (192MB L2).
// ---------------------------------------------------------------------------
__global__ __launch_bounds__(256) void layer_gemm_kernel(
    const float* __restrict__ h_in, const float* __restrict__ W,
    const float* __restrict__ t, float* __restrict__ h_out) {
  const int wave = threadIdx.x >> 5;
  const int lane = threadIdx.x & 31;
  const int half = lane >> 4;   // K-half selector for A/B frags
  const int l15  = lane & 15;   // M index (A) / N index (B,C)
  const int m0 = blockIdx.x * 128 + wave * 16;
  const int o0 = blockIdx.y * 64;

  const float* aRow = h_in + (size_t)(m0 + l15) * HID + 2 * half;
  const float* bRow = W    + (size_t)(o0 + l15) * HID + 2 * half;

  v8f acc[4] = {};
  for (int k = 0; k < HID; k += 4) {
    v2f a = *(const v2f*)(aRow + k);  // A[m=l15][k + 2*half + {0,1}]
#pragma unroll
    for (int bj = 0; bj < 4; ++bj) {
      v2f b = *(const v2f*)(bRow + (size_t)(bj * 16) * HID + k);
      acc[bj] = __builtin_amdgcn_wmma_f32_16x16x4_f32(
          false, a, false, b, (short)0, acc[bj], false, false);
    }
  }

  const float scaleN = (float)NROWS;
  const float invNm1 = 1.0f / (float)(NROWS - 1);
#pragma unroll
  for (int bj = 0; bj < 4; ++bj) {
    const int col  = o0 + bj * 16 + l15;
    const float tc = t[col];
#pragma unroll
    for (int r = 0; r < 8; ++r) {
      const int row = m0 + r + 8 * half;  // C/D layout: M = r + 8*half
      const float f = (scaleN * acc[bj][r] - tc) * invNm1;
      h_out[(size_t)row * HID + col] = h_in[(size_t)row * HID + col] + f;
    }
  }
}

// ---------------------------------------------------------------------------
// Stage 4: in-place  d_out[l] <- sigmoid(d_out[l] @ W_ff^T + b_ff).
// Each block owns 16 full rows: stage them in LDS (padded stride), barrier,
// then 8 waves each compute a 16x64 WMMA tile and overwrite the rows.
// ---------------------------------------------------------------------------
__global__ __launch_bounds__(256) void ff_sigmoid_kernel(
    float* __restrict__ io, const float* __restrict__ Wff,
    const float* __restrict__ bff) {
  __shared__ float lds[16 * LDS_STRIDE];
  const int l  = blockIdx.y;
  const int m0 = blockIdx.x * 16;
  float* base = io + (size_t)l * ((size_t)NROWS * HID) + (size_t)m0 * HID;

  const int t = threadIdx.x;
#pragma unroll
  for (int i = 0; i < 32; ++i) {            // 16*512 floats / 256 threads
    const int idx = t + i * 256;
    const int row = idx >> 9, col = idx & 511;
    lds[row * LDS_STRIDE + col] = base[(size_t)row * HID + col];
  }
  __syncthreads();

  const int wave = t >> 5;
  const int lane = t & 31;
  const int half = lane >> 4;
  const int l15  = lane & 15;
  const int o0 = wave * 64;

  const float* aRow = &lds[l15 * LDS_STRIDE + 2 * half];
  const float* bRow = Wff + (size_t)(o0 + l15) * HID + 2 * half;

  v8f acc[4] = {};
  for (int k = 0; k < HID; k += 4) {
    v2f a = *(const v2f*)(aRow + k);        // ds_load_b64, conflict-free
#pragma unroll
    for (int bj = 0; bj < 4; ++bj) {
      v2f b = *(const v2f*)(bRow + (size_t)(bj * 16) * HID + k);
      acc[bj] = __builtin_amdgcn_wmma_f32_16x16x4_f32(
          false, a, false, b, (short)0, acc[bj], false, false);
    }
  }

#pragma unroll
  for (int bj = 0; bj < 4; ++bj) {
    const int col = o0 + bj * 16 + l15;
    const float bv = bff[col];
#pragma unroll
    for (int r = 0; r < 8; ++r) {
      const float v = acc[bj][r] + bv;
      base[(size_t)(r + 8 * half) * HID + col] = 1.0f / (1.0f + expf(-v));
    }
  }
}

// ---------------------------------------------------------------------------
extern "C" void kernel_launch(void* const* d_in, const int* in_sizes, int n_in,
                              void* d_out, int out_size, void* d_ws,
                              size_t ws_size, hipStream_t stream) {
  (void)in_sizes; (void)n_in; (void)out_size; (void)ws_size;
  const float* x   = (const float*)d_in[0];   // [1, N, H]
  const float* Ws  = (const float*)d_in[1];   // [L, H, H] (out, in)
  const float* Wff = (const float*)d_in[2];   // [H, H]    (out, in)
  const float* bff = (const float*)d_in[3];   // [H]
  float* out = (float*)d_out;                 // [L, N, H] — also holds all_h

  float* partial = (float*)d_ws;                       // 256*512 floats
  float* tvec    = partial + (size_t)CS_BLOCKS * HID;  // 512 floats

  const size_t NH = (size_t)NROWS * HID;
  for (int l = 0; l < LAYERS; ++l) {
    const float* h_in = (l == 0) ? x : out + (size_t)(l - 1) * NH;
    const float* Wl   = Ws + (size_t)l * HID * HID;
    float* h_out      = out + (size_t)l * NH;

    colsum_partial_kernel<<<CS_BLOCKS, 256, 0, stream>>>(h_in, partial);
    reduce_t_kernel<<<1, 512, 0, stream>>>(partial, Wl, tvec);
    layer_gemm_kernel<<<dim3(NROWS / 128, HID / 64), 256, 0, stream>>>(
        h_in, Wl, tvec, h_out);
  }
  ff_sigmoid_kernel<<<dim3(NROWS / 16, LAYERS), 256, 0, stream>>>(out, Wff, bff);
}